// OSSBlock_83915071029771
// MI455X (gfx1250) — compile-verified
//
#include <hip/hip_runtime.h>
#include <math.h>

typedef float v2f __attribute__((ext_vector_type(2)));
typedef float v8f __attribute__((ext_vector_type(8)));

#define BB_  4
#define C_   64
#define H_   64
#define W_   64
#define HW_  (H_*W_)      // 4096
#define L_   HW_
#define DI_  512          // m1 d_inner
#define XZC_ 1024         // m1 2*d_inner
#define DSTATE_ 16
#define DTR_ 16           // m1 dt_rank
#define XDBL_ 48          // dtr + 2*16

__device__ __forceinline__ float siluf(float x){ return x / (1.0f + __expf(-x)); }
__device__ __forceinline__ float softplusf(float x){ return fmaxf(x, 0.0f) + log1pf(__expf(-fabsf(x))); }
__device__ __forceinline__ float geluf(float x){ return 0.5f * x * (1.0f + erff(x * 0.70710678118654752440f)); }

// ---------------- LayerNorm over (C,H,W) per batch ----------------
__global__ void ln_kernel(const float* __restrict__ x, const float* __restrict__ w,
                          const float* __restrict__ b, float* __restrict__ out, int n){
  int bb = blockIdx.x;
  const float* xb = x + (long long)bb * n;
  float* ob = out + (long long)bb * n;
  __shared__ float s1[256], s2[256];
  float a = 0.f, a2 = 0.f;
  for (int i = threadIdx.x; i < n; i += 256){ float v = xb[i]; a += v; a2 += v*v; }
  s1[threadIdx.x] = a; s2[threadIdx.x] = a2; __syncthreads();
  for (int off = 128; off > 0; off >>= 1){
    if (threadIdx.x < off){ s1[threadIdx.x] += s1[threadIdx.x+off]; s2[threadIdx.x] += s2[threadIdx.x+off]; }
    __syncthreads();
  }
  float mu = s1[0] / (float)n;
  float var = s2[0] / (float)n - mu*mu;
  float rs = rsqrtf(var + 1e-5f);
  for (int i = threadIdx.x; i < n; i += 256)
    ob[i] = (xb[i] - mu) * rs * w[i] + b[i];
}

// ---------------- Generic strided GEMM via V_WMMA_F32_16X16X4_F32 ----------------
// C[m,n] = act( sum_k A[m,k]*B[k,n] + bias_m[m] + bias_n[n] ) + add[m,n]
// act: 0=none 1=silu 2=softplus.
// One 16(m) x 32(n) output per wave (two accumulators share each A fragment);
// 8 waves/block arranged 2(m) x 4(n) -> block covers 32m x 128n.
// AK1/BK1: compile-time "K-stride == 1" specialization (contiguous v2f loads,
// pointer-increment addressing, no per-iteration branches).
template<bool AK1, bool BK1>
__global__ void gemm_wmma_kernel(
    const float* __restrict__ A, long long sa_m, long long sa_k, long long sa_b,
    const float* __restrict__ Bm, long long sb_k, long long sb_n, long long sb_b,
    float* __restrict__ Cm, long long sc_m, long long sc_b,
    int M, int N, int K,
    const float* __restrict__ bias_m, const float* __restrict__ bias_n,
    const float* __restrict__ addp, int act)
{
  int bb = blockIdx.z;
  A  += (long long)bb * sa_b;
  Bm += (long long)bb * sb_b;
  Cm += (long long)bb * sc_b;
  const float* addb = addp ? (addp + (long long)bb * sc_b) : nullptr;

  int wave = threadIdx.x >> 5, lane = threadIdx.x & 31;
  int wm = wave >> 2, wn = wave & 3;
  int tm0 = (blockIdx.y * 2 + wm) * 16;
  int tn0 = (blockIdx.x * 4 + wn) * 32;
  if (tm0 >= M || tn0 >= N) return;   // whole-wave skip: WMMA always runs with full EXEC

  int mr = lane & 15, kh = lane >> 4;
  int row  = tm0 + mr;       if (row  > M-1) row  = M-1;
  int col0 = tn0 + mr;       if (col0 > N-1) col0 = N-1;
  int col1 = tn0 + 16 + mr;  if (col1 > N-1) col1 = N-1;

  int ka0 = 2*kh;   // A 16x4 layout: lanes0-15 K={0,1}, lanes16-31 K={2,3}; B assumed symmetric
  const float* pa  = A  + (long long)row  * sa_m + (AK1 ? (long long)ka0 : (long long)ka0 * sa_k);
  const float* pb0 = Bm + (long long)col0 * sb_n + (BK1 ? (long long)ka0 : (long long)ka0 * sb_k);
  const float* pb1 = Bm + (long long)col1 * sb_n + (BK1 ? (long long)ka0 : (long long)ka0 * sb_k);
  const long long stepA = AK1 ? 4 : 4 * sa_k;
  const long long stepB = BK1 ? 4 : 4 * sb_k;

  v8f acc0 = {0.f,0.f,0.f,0.f,0.f,0.f,0.f,0.f};
  v8f acc1 = {0.f,0.f,0.f,0.f,0.f,0.f,0.f,0.f};

  auto ldA  = [&]() -> v2f { if (AK1) return *(const v2f*)pa;
                             v2f r; r.x = pa[0];  r.y = pa[sa_k];  return r; };
  auto ldB0 = [&]() -> v2f { if (BK1) return *(const v2f*)pb0;
                             v2f r; r.x = pb0[0]; r.y = pb0[sb_k]; return r; };
  auto ldB1 = [&]() -> v2f { if (BK1) return *(const v2f*)pb1;
                             v2f r; r.x = pb1[0]; r.y = pb1[sb_k]; return r; };

  int k0 = 0;
  // main pipeline: 64-K chunks, 16 fully-unrolled steps, 2 WMMAs per A fragment
  for (; k0 + 64 <= K; k0 += 64){
    if (k0 + 64 < K){
      __builtin_prefetch(pa  + 16*stepA, 0, 1);   // global_prefetch_b8: next A chunk
      __builtin_prefetch(pb0 + 16*stepB, 0, 1);
    }
    #pragma unroll
    for (int u = 0; u < 16; u++){
      v2f av = ldA(), b0 = ldB0(), b1 = ldB1();
      acc0 = __builtin_amdgcn_wmma_f32_16x16x4_f32(false, av, false, b0, (short)0, acc0, false, false);
      acc1 = __builtin_amdgcn_wmma_f32_16x16x4_f32(false, av, false, b1, (short)0, acc1, false, false);
      pa += stepA; pb0 += stepB; pb1 += stepB;
    }
  }
  // remainder (K % 64, still multiple of 4)
  for (; k0 < K; k0 += 4){
    v2f av = ldA(), b0 = ldB0(), b1 = ldB1();
    acc0 = __builtin_amdgcn_wmma_f32_16x16x4_f32(false, av, false, b0, (short)0, acc0, false, false);
    acc1 = __builtin_amdgcn_wmma_f32_16x16x4_f32(false, av, false, b1, (short)0, acc1, false, false);
    pa += stepA; pb0 += stepB; pb1 += stepB;
  }

  #pragma unroll
  for (int r = 0; r < 8; r++){
    int m = tm0 + r + 8*kh;             // C/D layout: vgpr r, lanes0-15 -> M=r, lanes16-31 -> M=8+r
    if (m >= M) continue;
    #pragma unroll
    for (int t = 0; t < 2; t++){
      int n = tn0 + t*16 + mr;
      if (n >= N) continue;
      float v = (t == 0) ? acc0[r] : acc1[r];
      if (bias_m) v += bias_m[m];
      if (bias_n) v += bias_n[n];
      if (act == 1) v = siluf(v);
      else if (act == 2) v = softplusf(v);
      long long ci = (long long)m * sc_m + n;
      if (addb) v += addb[ci];
      Cm[ci] = v;
    }
  }
}

// ---------------- depthwise 3x3 (pad 1), optional SiLU ----------------
__global__ void dwconv3_kernel(const float* __restrict__ in, long long bs_in,
                               const float* __restrict__ w, const float* __restrict__ bias,
                               float* __restrict__ out, long long bs_out,
                               int C, int act, int total){
  int idx = blockIdx.x * 256 + threadIdx.x;
  if (idx >= total) return;
  int xx = idx % W_; int yy = (idx / W_) % H_; int c = (idx / HW_) % C; int b = idx / (HW_ * C);
  const float* ip = in + (long long)b * bs_in + (long long)c * HW_;
  const float* wc = w + c * 9;
  float acc = bias ? bias[c] : 0.f;
  #pragma unroll
  for (int ky = 0; ky < 3; ky++){
    int y2 = yy + ky - 1; if (y2 < 0 || y2 >= H_) continue;
    #pragma unroll
    for (int kx = 0; kx < 3; kx++){
      int x2 = xx + kx - 1; if (x2 < 0 || x2 >= W_) continue;
      acc += wc[ky*3+kx] * ip[y2*W_ + x2];
    }
  }
  if (act) acc = siluf(acc);
  out[(long long)b * bs_out + (long long)c * HW_ + yy*W_ + xx] = acc;
}

// ---------------- strided SiLU (fo2 gate) ----------------
__global__ void silu_strided_kernel(const float* __restrict__ in, float* __restrict__ out,
                                    long long bs, int nper, int total){
  int idx = blockIdx.x * 256 + threadIdx.x;
  if (idx >= total) return;
  int b = idx / nper, r = idx % nper;
  out[(long long)b*bs + r] = siluf(in[(long long)b*bs + r]);
}

// ---------------- build fop = [hf, hf_rev, wf, wf_rev] : (B, L, 256) ----------------
__global__ void build_fop_kernel(const float* __restrict__ f1, float* __restrict__ fop, int total){
  int idx = blockIdx.x * 256 + threadIdx.x;
  if (idx >= total) return;
  int cc = idx & 63; int l = (idx >> 6) % L_; int b = idx / (64 * L_);
  const float* fb = f1 + (long long)b * 64 * HW_ + (long long)cc * HW_;
  int lr = L_ - 1 - l;
  float hf0 = fb[l];
  float hf1 = fb[lr];
  float wf0 = fb[(l  % H_) * W_ + (l  / H_)];
  float wf1 = fb[(lr % H_) * W_ + (lr / H_)];
  float* o = fop + (long long)(b * L_ + l) * 256;
  o[cc] = hf0; o[64+cc] = hf1; o[128+cc] = wf0; o[192+cc] = wf1;
}

// ---------------- causal depthwise conv1d (k=4) + SiLU on xi channels of xz ----------------
__global__ void conv1d_silu_kernel(const float* __restrict__ xz, const float* __restrict__ w,
                                   const float* __restrict__ bias, float* __restrict__ xi, int total){
  int idx = blockIdx.x * 256 + threadIdx.x;
  if (idx >= total) return;
  int j = idx % DI_; int l = (idx / DI_) % L_; int b = idx / (DI_ * L_);
  float acc = bias[j];
  #pragma unroll
  for (int t = 0; t < 4; t++){
    int ll = l - 3 + t;
    if (ll >= 0) acc += w[j*4 + t] * xz[(long long)(b*L_ + ll) * XZC_ + j];
  }
  xi[(long long)(b*L_ + l) * DI_ + j] = siluf(acc);
}

// ---------------- selective scan (m1), fused D-skip + SiLU(z) gate ----------------
__global__ void scan_kernel(const float* __restrict__ dt, const float* __restrict__ xi,
                            const float* __restrict__ xdbl, const float* __restrict__ Alog,
                            const float* __restrict__ Dp, const float* __restrict__ xz,
                            float* __restrict__ yg){
  int b = blockIdx.x; int j = threadIdx.x;
  __shared__ float sBC[64 * 32];
  float As[DSTATE_], h[DSTATE_];
  #pragma unroll
  for (int s = 0; s < DSTATE_; s++){ As[s] = -__expf(Alog[j*DSTATE_ + s]); h[s] = 0.f; }
  float Dj = Dp[j];
  for (int l0 = 0; l0 < L_; l0 += 64){
    __syncthreads();
    for (int i = threadIdx.x; i < 64*32; i += 512){
      int st = i >> 5, q = i & 31;
      sBC[i] = xdbl[(long long)(b*L_ + l0 + st) * XDBL_ + DTR_ + q];  // B at [16,32), C at [32,48)
    }
    __syncthreads();
    for (int sl = 0; sl < 64; sl++){
      long long base = (long long)(b*L_ + l0 + sl);
      float dtv = dt[base*DI_ + j];
      float xiv = xi[base*DI_ + j];
      float dx  = dtv * xiv;
      float ys  = 0.f;
      const float* bc = &sBC[sl*32];
      #pragma unroll
      for (int s = 0; s < DSTATE_; s++){
        float dA = __expf(dtv * As[s]);
        h[s] = dA * h[s] + dx * bc[s];
        ys = fmaf(h[s], bc[16 + s], ys);
      }
      float zv = xz[base*XZC_ + DI_ + j];
      yg[base*DI_ + j] = (ys + xiv * Dj) * siluf(zv);
    }
  }
}

// ---------------- 4-direction sum + gate with fo2: res (B,64,HW) ----------------
__global__ void residual_kernel(const float* __restrict__ yo, const float* __restrict__ fo,
                                float* __restrict__ res, int total){
  int idx = blockIdx.x * 256 + threadIdx.x;
  if (idx >= total) return;
  int l = idx % HW_; int cc = (idx / HW_) & 63; int b = idx / (HW_ * 64);
  long long base = (long long)(b*L_ + l) * 256 + cc;
  float s = yo[base] + yo[base + 64] + yo[base + 128] + yo[base + 192];
  float f2 = fo[(long long)b * 128 * HW_ + (long long)(64 + cc) * HW_ + l];
  res[(long long)b * 64 * HW_ + (long long)cc * HW_ + l] = s * f2;
}

// ---------------- per-(b,c) spatial mean ----------------
__global__ void colmean_kernel(const float* __restrict__ res, float* __restrict__ foc){
  int bc = blockIdx.x;
  const float* p = res + (long long)bc * HW_;
  __shared__ float s[256];
  float a = 0.f;
  for (int i = threadIdx.x; i < HW_; i += 256) a += p[i];
  s[threadIdx.x] = a; __syncthreads();
  for (int off = 128; off > 0; off >>= 1){
    if (threadIdx.x < off) s[threadIdx.x] += s[threadIdx.x + off];
    __syncthreads();
  }
  if (threadIdx.x == 0) foc[bc] = s[0] * (1.0f / HW_);
}

// ---------------- tiny mamba2 (d_model=2, L=64) -> lamb[b*64+l] ----------------
__global__ void mamba2_kernel(const float* __restrict__ foc,
    const float* __restrict__ in_w, const float* __restrict__ conv_w, const float* __restrict__ conv_b,
    const float* __restrict__ xproj_w, const float* __restrict__ dt_w, const float* __restrict__ dt_b,
    const float* __restrict__ Alog, const float* __restrict__ Dp, const float* __restrict__ out_w,
    float* __restrict__ lamb){
  int b = blockIdx.x; int l = threadIdx.x;   // 64 threads
  __shared__ float sxz[64][8];
  __shared__ float sxi[64][4];
  __shared__ float sdt[64][4];
  __shared__ float sB[64][16];
  __shared__ float sC[64][16];
  __shared__ float sys[64][4];
  float u0 = foc[b*64 + l], u1 = foc[b*64 + 63 - l];
  #pragma unroll
  for (int jj = 0; jj < 8; jj++) sxz[l][jj] = u0 * in_w[jj*2 + 0] + u1 * in_w[jj*2 + 1];
  __syncthreads();
  #pragma unroll
  for (int j = 0; j < 4; j++){
    float a = conv_b[j];
    #pragma unroll
    for (int t = 0; t < 4; t++){ int ll = l - 3 + t; if (ll >= 0) a += conv_w[j*4 + t] * sxz[ll][j]; }
    sxi[l][j] = siluf(a);
  }
  float xd0 = 0.f;
  #pragma unroll
  for (int j = 0; j < 4; j++) xd0 += sxi[l][j] * xproj_w[j];      // row 0 (dt_rank=1)
  #pragma unroll
  for (int j = 0; j < 4; j++) sdt[l][j] = softplusf(xd0 * dt_w[j] + dt_b[j]);
  #pragma unroll
  for (int s = 0; s < 16; s++){
    float bq = 0.f, cq = 0.f;
    #pragma unroll
    for (int j = 0; j < 4; j++){
      bq += sxi[l][j] * xproj_w[(1 + s)*4 + j];
      cq += sxi[l][j] * xproj_w[(17 + s)*4 + j];
    }
    sB[l][s] = bq; sC[l][s] = cq;
  }
  __syncthreads();
  if (l == 0){
    float h[4][16];
    for (int j = 0; j < 4; j++) for (int s = 0; s < 16; s++) h[j][s] = 0.f;
    for (int t = 0; t < 64; t++){
      for (int j = 0; j < 4; j++){
        float dv = sdt[t][j], dx = dv * sxi[t][j], ys = 0.f;
        for (int s = 0; s < 16; s++){
          float Av = -__expf(Alog[j*16 + s]);
          h[j][s] = __expf(dv * Av) * h[j][s] + dx * sB[t][s];
          ys += h[j][s] * sC[t][s];
        }
        sys[t][j] = ys;
      }
    }
  }
  __syncthreads();
  float o0 = 0.f, o1 = 0.f;
  #pragma unroll
  for (int j = 0; j < 4; j++){
    float y = sys[l][j] + sxi[l][j] * Dp[j];
    y *= siluf(sxz[l][4 + j]);
    o0 += y * out_w[0*4 + j];
    o1 += y * out_w[1*4 + j];
  }
  lamb[b*64 + l] = o0 + o1;
}

// ---------------- foss = res * (1 + lamb) ----------------
__global__ void foss_kernel(const float* __restrict__ res, const float* __restrict__ lamb,
                            float* __restrict__ foss, int total){
  int idx = blockIdx.x * 256 + threadIdx.x;
  if (idx >= total) return;
  int cc = (idx / HW_) & 63; int b = idx / (HW_ * 64);
  foss[idx] = res[idx] * (1.0f + lamb[b*64 + cc]);
}

// ---------------- gg = gelu(g1) * g2 ----------------
__global__ void gelugate_kernel(const float* __restrict__ gd, float* __restrict__ gg, int total){
  int idx = blockIdx.x * 256 + threadIdx.x;
  if (idx >= total) return;
  int l = idx % HW_; int i = (idx / HW_) & 127; int b = idx / (HW_ * 128);
  float g1 = gd[(long long)b * 256 * HW_ + (long long)i        * HW_ + l];
  float g2 = gd[(long long)b * 256 * HW_ + (long long)(128 + i)* HW_ + l];
  gg[(long long)b * 128 * HW_ + (long long)i * HW_ + l] = geluf(g1) * g2;
}

// ============================ host ============================
static inline void launch_gemm(hipStream_t s,
    const float* A, long long sam, long long sak, long long sab,
    const float* Bp, long long sbk, long long sbn, long long sbb,
    float* Cp, long long scm, long long scb,
    int M, int N, int K, const float* bm, const float* bn, const float* add, int act, int batches){
  dim3 grid((unsigned)((N + 127) / 128), (unsigned)((M + 31) / 32), (unsigned)batches);
  bool ak1 = (sak == 1), bk1 = (sbk == 1);
  if (ak1 && bk1)
    gemm_wmma_kernel<true,true><<<grid, 256, 0, s>>>(A,sam,sak,sab,Bp,sbk,sbn,sbb,Cp,scm,scb,M,N,K,bm,bn,add,act);
  else if (ak1)
    gemm_wmma_kernel<true,false><<<grid, 256, 0, s>>>(A,sam,sak,sab,Bp,sbk,sbn,sbb,Cp,scm,scb,M,N,K,bm,bn,add,act);
  else if (bk1)
    gemm_wmma_kernel<false,true><<<grid, 256, 0, s>>>(A,sam,sak,sab,Bp,sbk,sbn,sbb,Cp,scm,scb,M,N,K,bm,bn,add,act);
  else
    gemm_wmma_kernel<false,false><<<grid, 256, 0, s>>>(A,sam,sak,sab,Bp,sbk,sbn,sbb,Cp,scm,scb,M,N,K,bm,bn,add,act);
}

extern "C" void kernel_launch(void* const* d_in, const int* in_sizes, int n_in,
                              void* d_out, int out_size, void* d_ws, size_t ws_size,
                              hipStream_t stream) {
  const float* x      = (const float*)d_in[0];
  const float* n1w    = (const float*)d_in[1];
  const float* n1b    = (const float*)d_in[2];
  const float* n2w    = (const float*)d_in[3];
  const float* n2b    = (const float*)d_in[4];
  const float* ipw    = (const float*)d_in[5];
  const float* ipb    = (const float*)d_in[6];
  const float* dww    = (const float*)d_in[7];
  const float* dwb    = (const float*)d_in[8];
  const float* opw    = (const float*)d_in[9];
  const float* opb    = (const float*)d_in[10];
  const float* m1_in  = (const float*)d_in[11];
  const float* m1_cw  = (const float*)d_in[12];
  const float* m1_cb  = (const float*)d_in[13];
  const float* m1_xp  = (const float*)d_in[14];
  const float* m1_dtw = (const float*)d_in[15];
  const float* m1_dtb = (const float*)d_in[16];
  const float* m1_Al  = (const float*)d_in[17];
  const float* m1_D   = (const float*)d_in[18];
  const float* m1_ow  = (const float*)d_in[19];
  const float* m2_in  = (const float*)d_in[20];
  const float* m2_cw  = (const float*)d_in[21];
  const float* m2_cb  = (const float*)d_in[22];
  const float* m2_xp  = (const float*)d_in[23];
  const float* m2_dtw = (const float*)d_in[24];
  const float* m2_dtb = (const float*)d_in[25];
  const float* m2_Al  = (const float*)d_in[26];
  const float* m2_D   = (const float*)d_in[27];
  const float* m2_ow  = (const float*)d_in[28];
  const float* ffn_iw = (const float*)d_in[29];
  const float* ffn_ib = (const float*)d_in[30];
  const float* ffn_dw = (const float*)d_in[31];
  const float* ffn_db = (const float*)d_in[32];
  const float* ffn_ow = (const float*)d_in[33];
  const float* ffn_ob = (const float*)d_in[34];
  float* out = (float*)d_out;

  // ---- workspace carve (floats) ----
  float* f    = (float*)d_ws;
  float* xn   = f;                      // 1,048,576   (B,64,HW); reused as xn2 later
  float* fo   = xn   + 1048576;         // 2,097,152   (B,128,HW)
  float* fo1s = fo   + 2097152;         // 1,048,576   (B,64,HW)
  float* fop  = fo1s + 1048576;         // 4,194,304   (B,L,256)
  float* xz   = fop  + 4194304;         // 16,777,216  (B,L,1024); later: g + gd aliases
  float* xi   = xz   + 16777216;        // 8,388,608   (B,L,512);  later: gg alias
  float* xdbl = xi   + 8388608;         //   786,432   (B,L,48)
  float* dtb  = xdbl + 786432;          // 8,388,608   (B,L,512)
  float* yg   = dtb  + 8388608;         // 8,388,608   (B,L,512)
  float* yo   = yg   + 8388608;         // 4,194,304   (B,L,256)
  float* res  = yo   + 4194304;         // 1,048,576   (B,64,HW)
  float* foss = res  + 1048576;         // 1,048,576
  float* x2   = foss + 1048576;         // 1,048,576
  float* foc  = x2   + 1048576;         // 256
  float* lamb = foc  + 256;             // 256
  float* g    = xz;                     // alias (B,256,HW)
  float* gd   = xz + 4194304;           // alias (B,256,HW)
  float* gg   = xi;                     // alias (B,128,HW)

  const long long HW = HW_;

  // 1) LayerNorm 1
  ln_kernel<<<BB_, 256, 0, stream>>>(x, n1w, n1b, xn, C_*HW_);

  // 2) inproj 1x1: fo(b,128,HW) = W(128,64) @ xn(b,64,HW) + b
  launch_gemm(stream, ipw, 64, 1, 0, xn, HW, 1, 64*HW, fo, HW, 128*HW,
              128, HW_, 64, ipb, nullptr, nullptr, 0, BB_);

  // 3) fo1 -> dwconv3 + SiLU
  dwconv3_kernel<<<(BB_*64*HW_+255)/256, 256, 0, stream>>>(fo, 128*HW, dww, dwb,
                                                           fo1s, 64*HW, 64, 1, BB_*64*HW_);
  // 4) fo2 -> SiLU (in place)
  silu_strided_kernel<<<(BB_*64*HW_+255)/256, 256, 0, stream>>>(fo + 64*HW_, fo + 64*HW_,
                                                                128*HW, 64*HW_, BB_*64*HW_);
  // 5) pack 4 scan directions
  build_fop_kernel<<<(BB_*L_*64+255)/256, 256, 0, stream>>>(fo1s, fop, BB_*L_*64);

  // 6) m1 in-proj: xz(BL,1024) = fop(BL,256) @ in_w^T
  launch_gemm(stream, fop, 256, 1, 0, m1_in, 1, 256, 0, xz, XZC_, 0,
              BB_*L_, XZC_, 256, nullptr, nullptr, nullptr, 0, 1);

  // 7) causal conv1d + SiLU -> xi(BL,512)
  conv1d_silu_kernel<<<(BB_*L_*DI_+255)/256, 256, 0, stream>>>(xz, m1_cw, m1_cb, xi, BB_*L_*DI_);

  // 8) x_dbl(BL,48) = xi @ xproj^T
  launch_gemm(stream, xi, DI_, 1, 0, m1_xp, 1, DI_, 0, xdbl, XDBL_, 0,
              BB_*L_, XDBL_, DI_, nullptr, nullptr, nullptr, 0, 1);

  // 9) dt(BL,512) = softplus(x_dbl[:, :16] @ dt_w^T + dt_b)
  launch_gemm(stream, xdbl, XDBL_, 1, 0, m1_dtw, 1, DTR_, 0, dtb, DI_, 0,
              BB_*L_, DI_, DTR_, nullptr, m1_dtb, nullptr, 2, 1);

  // 10) selective scan (fused D-skip + SiLU(z) gate) -> yg(BL,512)
  scan_kernel<<<BB_, DI_, 0, stream>>>(dtb, xi, xdbl, m1_Al, m1_D, xz, yg);

  // 11) m1 out-proj: yo(BL,256) = yg @ out_w^T
  launch_gemm(stream, yg, DI_, 1, 0, m1_ow, 1, DI_, 0, yo, 256, 0,
              BB_*L_, 256, DI_, nullptr, nullptr, nullptr, 0, 1);

  // 12) sum 4 directions, gate with SiLU(fo2) -> res(b,64,HW)
  residual_kernel<<<(BB_*64*HW_+255)/256, 256, 0, stream>>>(yo, fo, res, BB_*64*HW_);

  // 13) channel means
  colmean_kernel<<<BB_*64, 256, 0, stream>>>(res, foc);

  // 14) tiny mamba2 -> lamb
  mamba2_kernel<<<BB_, 64, 0, stream>>>(foc, m2_in, m2_cw, m2_cb, m2_xp, m2_dtw, m2_dtb,
                                        m2_Al, m2_D, m2_ow, lamb);

  // 15) foss = res * (1 + lamb)
  foss_kernel<<<(BB_*64*HW_+255)/256, 256, 0, stream>>>(res, lamb, foss, BB_*64*HW_);

  // 16) out-proj 1x1 + residual x -> x2
  launch_gemm(stream, opw, 64, 1, 0, foss, HW, 1, 64*HW, x2, HW, 64*HW,
              64, HW_, 64, opb, nullptr, x, 0, BB_);

  // 17) LayerNorm 2 (reuse xn)
  ln_kernel<<<BB_, 256, 0, stream>>>(x2, n2w, n2b, xn, C_*HW_);

  // 18) ffn_in 1x1: g(b,256,HW)
  launch_gemm(stream, ffn_iw, 64, 1, 0, xn, HW, 1, 64*HW, g, HW, 256*HW,
              256, HW_, 64, ffn_ib, nullptr, nullptr, 0, BB_);

  // 19) dwconv3 on 256 channels
  dwconv3_kernel<<<(BB_*256*HW_+255)/256, 256, 0, stream>>>(g, 256*HW, ffn_dw, ffn_db,
                                                            gd, 256*HW, 256, 0, BB_*256*HW_);
  // 20) gelu gate -> gg(b,128,HW)
  gelugate_kernel<<<(BB_*128*HW_+255)/256, 256, 0, stream>>>(gd, gg, BB_*128*HW_);

  // 21) ffn_out 1x1 + residual x2 -> out
  launch_gemm(stream, ffn_ow, 128, 1, 0, gg, HW, 1, 128*HW, out, HW, 64*HW,
              64, HW_, 128, ffn_ob, nullptr, x2, 0, BB_);
}